// CumulativeLayerNorm_17660905521288
// MI455X (gfx1250) — compile-verified
//
#include <hip/hip_runtime.h>

// Problem constants (fixed by the harness / reference setup_inputs()).
#define B_  4
#define C_  512
#define T_  32000

typedef __attribute__((ext_vector_type(2))) float v2f;
typedef __attribute__((ext_vector_type(4))) float v4f;
typedef __attribute__((ext_vector_type(8))) float v8f;

#if defined(__gfx1250__) && __has_builtin(__builtin_amdgcn_wmma_f32_16x16x4_f32)
#define HAVE_WMMA_F32X4 1
#else
#define HAVE_WMMA_F32X4 0
#endif

// ---------------------------------------------------------------------------
// Kernel 1: per-(b,t) channel reduction producing s[b,t] = sum_c x[b,c,t] and
// sq[b,t] = sum_c x^2. Each wave owns a 16-wide time tile and uses
// V_WMMA_F32_16X16X4_F32 with an all-ones A matrix: D[m,n] = sum_k B[k,n],
// i.e. column sums of a 4-channel x 16-time f32 tile. Since A is all ones,
// any assignment of the 4 channel values into a column's K slots yields the
// same column sum, so only lane%16 <-> time-column matters (satisfied).
// D layout: VGPR0 lanes 0-15 hold row M=0 => the 16 column sums.
// Regular (temporal) loads on purpose: we want x to linger in the 192 MB L2
// so pass 3 (reverse order) can re-read the tail for free.
// ---------------------------------------------------------------------------
__global__ void __launch_bounds__(256)
cln_reduce_kernel(const float* __restrict__ x,
                  float* __restrict__ s_out,
                  float* __restrict__ sq_out) {
    const int tid  = threadIdx.x;
    const int wave = tid >> 5;
    const int lane = tid & 31;
    const int b    = blockIdx.y;
    const int t0   = (blockIdx.x * 8 + wave) << 4;   // 16 time steps per wave
    const float* xb = x + (size_t)b * C_ * T_;

#if HAVE_WMMA_F32X4
    const int half = lane >> 4;      // which pair of channels this lane feeds
    const int n    = lane & 15;      // time column within the tile
    const float* p = xb + (size_t)(2 * half) * T_ + (t0 + n);

    v8f acc  = {0.f, 0.f, 0.f, 0.f, 0.f, 0.f, 0.f, 0.f};
    v8f accq = {0.f, 0.f, 0.f, 0.f, 0.f, 0.f, 0.f, 0.f};
    const v2f ones = {1.0f, 1.0f};   // A (16x4) = all ones -> column sums

    for (int c0 = 0; c0 < C_; c0 += 4) {
        float b0 = p[0];
        float b1 = p[T_];
        p += (size_t)4 * T_;
        v2f bv = {b0, b1};
        v2f bq = {b0 * b0, b1 * b1};
        acc  = __builtin_amdgcn_wmma_f32_16x16x4_f32(
                   false, ones, false, bv, (short)0, acc,  false, false);
        accq = __builtin_amdgcn_wmma_f32_16x16x4_f32(
                   false, ones, false, bq, (short)0, accq, false, false);
    }
    if (lane < 16) {
        s_out [(size_t)b * T_ + t0 + lane] = acc[0];   // D row M=0, N=lane
        sq_out[(size_t)b * T_ + t0 + lane] = accq[0];
    }
#else
    // VALU fallback (kept compile-safe if the f32 WMMA builtin is absent).
    const int n = lane & 15;
    float s = 0.f, sq = 0.f;
    const float* p = xb + (t0 + n);
    for (int c = 0; c < C_; ++c) {
        float v = p[(size_t)c * T_];
        s  += v;
        sq += v * v;
    }
    if (lane < 16) {
        s_out [(size_t)b * T_ + t0 + n] = s;
        sq_out[(size_t)b * T_ + t0 + n] = sq;
    }
#endif
}

// ---------------------------------------------------------------------------
// Kernel 2: per-batch inclusive scan over T of (s, sq) -> (mean, rstd).
// One block per batch. Three-phase scan to keep the serial dependency chain
// short (the old chunked version had 125 dependent load round-trips):
//   A) 250 threads each sum a contiguous 128-element (float4-aligned) segment
//      -> all loads independent, fully pipelined, L2-resident from pass 1.
//   B) one 256-wide Hillis-Steele LDS scan of the 250 partials.
//   C) each thread replays its segment with its exclusive prefix, computing
//      mean/rstd and writing them as float4 stores.
// ---------------------------------------------------------------------------
__global__ void __launch_bounds__(256)
cln_scan_kernel(const float* __restrict__ s_in,
                const float* __restrict__ sq_in,
                float* __restrict__ mean_out,
                float* __restrict__ rstd_out) {
    __shared__ float px[2][256];
    __shared__ float py[2][256];

    const int tid = threadIdx.x;
    const int b   = blockIdx.x;
    const size_t base = (size_t)b * T_;

    const int  SEG    = 128;          // 250 * 128 == 32000
    const bool active = (tid < 250);
    const int  seg0   = tid * SEG;

    // Phase A: per-thread segment totals.
    float sx = 0.f, sy = 0.f;
    if (active) {
        const v4f* ps = (const v4f*)(s_in  + base + seg0);
        const v4f* pq = (const v4f*)(sq_in + base + seg0);
        for (int i = 0; i < SEG / 4; ++i) {
            v4f a = ps[i];
            v4f q = pq[i];
            sx += a.x + a.y + a.z + a.w;
            sy += q.x + q.y + q.z + q.w;
        }
    }

    // Phase B: block-wide inclusive scan of partials (inactive lanes add 0).
    int cur = 0;
    px[0][tid] = sx;
    py[0][tid] = sy;
    __syncthreads();
    for (int off = 1; off < 256; off <<= 1) {
        float ax = px[cur][tid];
        float ay = py[cur][tid];
        if (tid >= off) {
            ax += px[cur][tid - off];
            ay += py[cur][tid - off];
        }
        px[cur ^ 1][tid] = ax;
        py[cur ^ 1][tid] = ay;
        __syncthreads();
        cur ^= 1;
    }
    const float ex = px[cur][tid] - sx;   // exclusive prefix for this segment
    const float ey = py[cur][tid] - sy;

    // Phase C: replay segment with running prefix; emit mean / rstd.
    if (active) {
        const v4f* ps = (const v4f*)(s_in  + base + seg0);
        const v4f* pq = (const v4f*)(sq_in + base + seg0);
        v4f* pm = (v4f*)(mean_out + base + seg0);
        v4f* pr = (v4f*)(rstd_out + base + seg0);

        float rx = ex, ry = ey;
        for (int i = 0; i < SEG / 4; ++i) {
            v4f a = ps[i];
            v4f q = pq[i];
            v4f m, r;
            int t = seg0 + 4 * i;
            #pragma unroll
            for (int k = 0; k < 4; ++k) {
                rx += (k == 0 ? a.x : k == 1 ? a.y : k == 2 ? a.z : a.w);
                ry += (k == 0 ? q.x : k == 1 ? q.y : k == 2 ? q.z : q.w);
                const float count = (float)(t + k + 1) * (float)C_;
                const float mean  = rx / count;
                float var = ry / count - mean * mean;
                var = var < 0.f ? 0.f : var;
                const float rs = rsqrtf(var + 1e-8f);
                if (k == 0) { m.x = mean; r.x = rs; }
                else if (k == 1) { m.y = mean; r.y = rs; }
                else if (k == 2) { m.z = mean; r.z = rs; }
                else { m.w = mean; r.w = rs; }
            }
            pm[i] = m;
            pr[i] = r;
        }
    }
}

// ---------------------------------------------------------------------------
// Kernel 3: streaming normalize. y = (x - mean[b,t]) * rstd[b,t]*w[c] + bias[c]
//  - float4-vectorized along t (T % 4 == 0).
//  - Blocks traverse in REVERSE linearized order so the pass-1 L2-resident
//    tail of x (~192 MB of the 262 MB) is consumed before eviction.
//  - x is last-use and y is write-once -> non-temporal load/store hints keep
//    the 192 MB L2 available for the reused mean/rstd (1 MB, temporal).
//  - weight/bias are block-uniform -> scalar loads.
// ---------------------------------------------------------------------------
__global__ void __launch_bounds__(256)
cln_norm_kernel(const float* __restrict__ x,
                const float* __restrict__ weight,
                const float* __restrict__ bias,
                const float* __restrict__ mean,
                const float* __restrict__ rstd,
                float* __restrict__ y) {
    // Reverse block order: hit L2-resident tail of x first.
    const int bc  = (B_ * C_ - 1) - (int)blockIdx.x;
    const int seg = 7 - (int)blockIdx.y;
    const int b   = bc / C_;
    const int c   = bc % C_;

    const v4f* xp = (const v4f*)(x + ((size_t)b * C_ + c) * T_);
    v4f*       yp = (v4f*)(y + ((size_t)b * C_ + c) * T_);
    const v4f* mp = (const v4f*)(mean + (size_t)b * T_);
    const v4f* rp = (const v4f*)(rstd + (size_t)b * T_);

    const float wc = weight[c];
    const float bs = bias[c];

    const int seg_len = (T_ / 4) / 8;          // 1000 float4 per segment
    const int start   = seg * seg_len;
    const int end     = start + seg_len;

    for (int i = start + (int)threadIdx.x; i < end; i += 256) {
        v4f xv = __builtin_nontemporal_load(xp + i);   // last use of x
        v4f m  = mp[i];                                // temporal: reused 512x
        v4f r  = rp[i];
        v4f o;
        o.x = (xv.x - m.x) * r.x * wc + bs;
        o.y = (xv.y - m.y) * r.y * wc + bs;
        o.z = (xv.z - m.z) * r.z * wc + bs;
        o.w = (xv.w - m.w) * r.w * wc + bs;
        __builtin_nontemporal_store(o, yp + i);        // write-once output
    }
}

// ---------------------------------------------------------------------------
// Launch: workspace layout (floats): s[B*T] | sq[B*T] | mean[B*T] | rstd[B*T]
// = 4 * 4 * 32000 * 4 B = 2 MB.
// ---------------------------------------------------------------------------
extern "C" void kernel_launch(void* const* d_in, const int* in_sizes, int n_in,
                              void* d_out, int out_size, void* d_ws, size_t ws_size,
                              hipStream_t stream) {
    (void)in_sizes; (void)n_in; (void)out_size; (void)ws_size;

    const float* x      = (const float*)d_in[0];
    const float* weight = (const float*)d_in[1];
    const float* bias   = (const float*)d_in[2];
    float*       y      = (float*)d_out;

    float* ws   = (float*)d_ws;
    const size_t bt = (size_t)B_ * T_;
    float* s    = ws;
    float* sq   = ws + bt;
    float* mean = ws + 2 * bt;
    float* rstd = ws + 3 * bt;

    // Pass 1: channel reduction (WMMA column sums). 128 t per block.
    cln_reduce_kernel<<<dim3(T_ / 128, B_), 256, 0, stream>>>(x, s, sq);

    // Pass 2: per-batch 3-phase scan -> mean / rstd (latency-optimized).
    cln_scan_kernel<<<dim3(B_), 256, 0, stream>>>(s, sq, mean, rstd);

    // Pass 3: streaming normalize, reverse order for L2 tail reuse.
    cln_norm_kernel<<<dim3(B_ * C_, 8), 256, 0, stream>>>(x, weight, bias, mean, rstd, y);
}